// GCN_51531017617486
// MI455X (gfx1250) — compile-verified
//
#include <hip/hip_runtime.h>
#include <hip/hip_bf16.h>
#include <math.h>

// ---------------------------------------------------------------------------
// GATv2 VGAE encoder for MI455X (gfx1250, wave32).
//   layer1: xl1 = x@W1l, xr1 = x@W1r   (50000x128 @ 128x128, fp32 WMMA)
//   edge attention (score / softmax / aggregate) -> h, BatchNorm+ReLU
//   layer2/3: h@W{2,3}{l,r} (128->64) + attention -> mu, logvar
// GEMMs use V_WMMA_F32_16X16X4_F32 (exact fp32 matrix path).
// Edge passes: one wave32 per edge => coalesced row gathers/scatters,
// accumulators live in L2 (192MB), atomics via global_atomic_add_f32.
// ---------------------------------------------------------------------------

typedef __attribute__((ext_vector_type(2))) float v2f;
typedef __attribute__((ext_vector_type(8))) float v8f;

#define F_DIM 128   // input / hidden feature width (K of every GEMM)

// ---------------- fp32 WMMA GEMM:  C[M x N] = A[M x 128] * W[128 x N] -------
// block = 256 threads = 8 waves; each wave owns one 16x16 C tile.
// A-frag (16x4 f32): lane L -> M = L%16, holds K = (L/16)*2 + {0,1}
// B-frag (4x16 f32): lane L -> N = L%16, holds K = (L/16)*2 + {0,1}
// C/D   (16x16 f32): lane L -> N = L%16, VGPR v -> M = v + (L/16)*8
template <int N>
__launch_bounds__(256)
__global__ void gemm_wmma_f32(const float* __restrict__ A,
                              const float* __restrict__ W,
                              float* __restrict__ C, int M) {
  constexpr int NT  = N / 16;   // n-tiles covered by one block
  constexpr int MPB = 8 / NT;   // m-tiles per block (8 waves)
  const int wv   = threadIdx.x >> 5;
  const int lane = threadIdx.x & 31;
  const int mt = blockIdx.x * MPB + wv / NT;
  const int m0 = mt * 16;
  if (m0 >= M) return;                       // wave-uniform; EXEC stays full
  const int n0   = (wv % NT) * 16;
  const int half = lane >> 4;                // 0: K pair {0,1}, 1: {2,3}
  const int l16  = lane & 15;
  const int kb   = half * 2;
  const int mrow = m0 + l16;
  const int ncol = n0 + l16;

  const float* ap = A + (size_t)mrow * F_DIM + kb;   // 8B aligned
  v8f acc = {};
#pragma unroll 4
  for (int k = 0; k < F_DIM; k += 4) {
    v2f a = *reinterpret_cast<const v2f*>(ap + k);
    v2f b;
    b.x = W[(size_t)(k + kb)     * N + ncol];
    b.y = W[(size_t)(k + kb + 1) * N + ncol];
    acc = __builtin_amdgcn_wmma_f32_16x16x4_f32(
        /*neg_a=*/false, a, /*neg_b=*/false, b,
        /*c_mod=*/(short)0, acc, /*reuse_a=*/false, /*reuse_b=*/false);
  }
  float* crow = C + (size_t)(m0 + half * 8) * N + ncol;
#pragma unroll
  for (int v = 0; v < 8; ++v) crow[(size_t)v * N] = acc[v];
}

// ---------------- init: out = bias (broadcast), mmax = -inf, denom = 0 ------
__global__ void init_layer(float* __restrict__ mmax, float* __restrict__ denom,
                           float* __restrict__ out, const float* __restrict__ bias,
                           int n, int D) {
  int i = blockIdx.x * blockDim.x + threadIdx.x;
  int total = n * D;
  if (i < total) out[i] = bias[i % D];
  if (i < n) { mmax[i] = -INFINITY; denom[i] = 0.f; }
}

// ---------------- float atomic-max via monotone int/uint bit trick ----------
__device__ __forceinline__ void atomicMaxF32(float* addr, float val) {
  if (val >= 0.f) atomicMax(reinterpret_cast<int*>(addr), __float_as_int(val));
  else            atomicMin(reinterpret_cast<unsigned int*>(addr), __float_as_uint(val));
}

// ---------------- pass 1: e = a . leakyrelu(xl[src] + xr[dst]); seg-max -----
template <int D>
__launch_bounds__(256)
__global__ void edge_scores(const float* __restrict__ xl, const float* __restrict__ xr,
                            const float* __restrict__ att, const int* __restrict__ ei,
                            int nE, int nTot, float* __restrict__ e_out,
                            float* __restrict__ mmax) {
  const int wid  = blockIdx.x * (blockDim.x >> 5) + (threadIdx.x >> 5);
  if (wid >= nTot) return;
  const int lane = threadIdx.x & 31;
  int s, d;
  if (wid < nE) { s = ei[wid]; d = ei[nE + wid]; } else { s = d = wid - nE; }
  float acc = 0.f;
#pragma unroll
  for (int j = 0; j < D / 32; ++j) {
    const int k = lane + 32 * j;
    float v = xl[(size_t)s * D + k] + xr[(size_t)d * D + k];
    v = (v > 0.f) ? v : 0.2f * v;                     // LeakyReLU(0.2)
    acc += v * att[k];
  }
#pragma unroll
  for (int off = 16; off > 0; off >>= 1) acc += __shfl_xor(acc, off, 32);
  if (lane == 0) {
    e_out[wid] = acc;
    atomicMaxF32(&mmax[d], acc);
  }
}

// ---------------- pass 2: ex = exp(e - max[dst]); seg-sum -------------------
__global__ void edge_exp(const int* __restrict__ ei, int nE, int nTot,
                         float* __restrict__ e_buf, const float* __restrict__ mmax,
                         float* __restrict__ denom) {
  int e = blockIdx.x * blockDim.x + threadIdx.x;
  if (e >= nTot) return;
  int d = (e < nE) ? ei[nE + e] : (e - nE);
  float ex = expf(e_buf[e] - mmax[d]);
  e_buf[e] = ex;
  atomicAdd(&denom[d], ex);
}

// ---------------- pass 3: out[dst] += (ex/denom[dst]) * xl[src] -------------
template <int D>
__launch_bounds__(256)
__global__ void edge_aggr(const float* __restrict__ xl, const int* __restrict__ ei,
                          int nE, int nTot, const float* __restrict__ e_buf,
                          const float* __restrict__ denom, float* __restrict__ out) {
  const int wid  = blockIdx.x * (blockDim.x >> 5) + (threadIdx.x >> 5);
  if (wid >= nTot) return;
  const int lane = threadIdx.x & 31;
  int s, d;
  if (wid < nE) { s = ei[wid]; d = ei[nE + wid]; } else { s = d = wid - nE; }
  const float alpha = e_buf[wid] / denom[d];
#pragma unroll
  for (int j = 0; j < D / 32; ++j) {
    const int k = lane + 32 * j;
    atomicAdd(&out[(size_t)d * D + k], alpha * xl[(size_t)s * D + k]);
  }
}

// ---------------- BatchNorm batch stats (biased var), one block per col ----
__global__ void bn_stats(const float* __restrict__ h, int n,
                         float* __restrict__ mean, float* __restrict__ var) {
  __shared__ float ss[256], sq[256];
  const int f = blockIdx.x;
  float s = 0.f, q = 0.f;
  for (int i = threadIdx.x; i < n; i += blockDim.x) {
    float v = h[(size_t)i * F_DIM + f];
    s += v; q += v * v;
  }
  ss[threadIdx.x] = s; sq[threadIdx.x] = q;
  __syncthreads();
  for (int off = 128; off > 0; off >>= 1) {
    if (threadIdx.x < off) {
      ss[threadIdx.x] += ss[threadIdx.x + off];
      sq[threadIdx.x] += sq[threadIdx.x + off];
    }
    __syncthreads();
  }
  if (threadIdx.x == 0) {
    float m = ss[0] / (float)n;
    mean[f] = m;
    var[f]  = sq[0] / (float)n - m * m;
  }
}

// ---------------- BN normalize + ReLU (in place) ----------------------------
__global__ void bn_apply_relu(float* __restrict__ h, const float* __restrict__ mean,
                              const float* __restrict__ var, const float* __restrict__ gamma,
                              const float* __restrict__ beta, int total) {
  int i = blockIdx.x * blockDim.x + threadIdx.x;
  if (i >= total) return;
  int f = i & (F_DIM - 1);
  float v = (h[i] - mean[f]) * rsqrtf(var[f] + 1e-5f) * gamma[f] + beta[f];
  h[i] = (v > 0.f) ? v : 0.f;
}

// ---------------------------------------------------------------------------
extern "C" void kernel_launch(void* const* d_in, const int* in_sizes, int n_in,
                              void* d_out, int out_size, void* d_ws, size_t ws_size,
                              hipStream_t stream) {
  const float* x    = (const float*)d_in[0];
  const int*   ei   = (const int*)d_in[1];   // [2, E] (src row, dst row), int32
  const float* W1l  = (const float*)d_in[2];
  const float* W1r  = (const float*)d_in[3];
  const float* a1   = (const float*)d_in[4];
  const float* b1   = (const float*)d_in[5];
  const float* gmma = (const float*)d_in[6];
  const float* beta = (const float*)d_in[7];
  const float* W2l  = (const float*)d_in[8];
  const float* W2r  = (const float*)d_in[9];
  const float* a2   = (const float*)d_in[10];
  const float* b2   = (const float*)d_in[11];
  const float* W3l  = (const float*)d_in[12];
  const float* W3r  = (const float*)d_in[13];
  const float* a3   = (const float*)d_in[14];
  const float* b3   = (const float*)d_in[15];

  const int N = in_sizes[0] / F_DIM;      // 50000 nodes
  const int E = in_sizes[1] / 2;          // 600000 edges
  const int T = E + N;                    // + self loops
  const int O = 64;

  // -------- workspace layout (floats) --------
  float* base = (float*)d_ws;
  float* xl1  = base;                          // N*128
  float* xr1  = xl1  + (size_t)N * F_DIM;      // N*128
  float* h    = xr1  + (size_t)N * F_DIM;      // N*128
  float* ebuf = h    + (size_t)N * F_DIM;      // T
  float* mmax = ebuf + T;                      // N
  float* den  = mmax + N;                      // N
  float* mean = den  + N;                      // 128
  float* var  = mean + F_DIM;                  // 128
  // recycled after layer 1:
  float* xl2 = xl1;  float* xr2 = xl1 + (size_t)N * O;
  float* xl3 = xr1;  float* xr3 = xr1 + (size_t)N * O;

  float* mu  = (float*)d_out;
  float* lv  = mu + (size_t)N * O;

  const int BLK       = 256;
  const int gridGemmH = N / 16;                // 3125, N%16==0
  const int gridGemmO = (N / 16 + 1) / 2;      // 2 m-tiles per block at N=64
  const int gridWaveE = (T + 7) / 8;           // wave-per-edge, 8 waves/block
  const int gridElemH = (N * F_DIM + BLK - 1) / BLK;
  const int gridElemO = (N * O + BLK - 1) / BLK;
  const int gridEdge  = (T + BLK - 1) / BLK;

  // ================= layer 1: GATv2(128 -> 128) =================
  gemm_wmma_f32<F_DIM><<<gridGemmH, BLK, 0, stream>>>(x, W1l, xl1, N);
  gemm_wmma_f32<F_DIM><<<gridGemmH, BLK, 0, stream>>>(x, W1r, xr1, N);
  init_layer<<<gridElemH, BLK, 0, stream>>>(mmax, den, h, b1, N, F_DIM);
  edge_scores<F_DIM><<<gridWaveE, BLK, 0, stream>>>(xl1, xr1, a1, ei, E, T, ebuf, mmax);
  edge_exp<<<gridEdge, BLK, 0, stream>>>(ei, E, T, ebuf, mmax, den);
  edge_aggr<F_DIM><<<gridWaveE, BLK, 0, stream>>>(xl1, ei, E, T, ebuf, den, h);

  // ================= BatchNorm + ReLU =================
  bn_stats<<<F_DIM, BLK, 0, stream>>>(h, N, mean, var);
  bn_apply_relu<<<gridElemH, BLK, 0, stream>>>(h, mean, var, gmma, beta, N * F_DIM);

  // ================= layer 2: mu head (128 -> 64) =================
  gemm_wmma_f32<64><<<gridGemmO, BLK, 0, stream>>>(h, W2l, xl2, N);
  gemm_wmma_f32<64><<<gridGemmO, BLK, 0, stream>>>(h, W2r, xr2, N);
  init_layer<<<gridElemO, BLK, 0, stream>>>(mmax, den, mu, b2, N, O);
  edge_scores<64><<<gridWaveE, BLK, 0, stream>>>(xl2, xr2, a2, ei, E, T, ebuf, mmax);
  edge_exp<<<gridEdge, BLK, 0, stream>>>(ei, E, T, ebuf, mmax, den);
  edge_aggr<64><<<gridWaveE, BLK, 0, stream>>>(xl2, ei, E, T, ebuf, den, mu);

  // ================= layer 3: logvar head (128 -> 64) =================
  gemm_wmma_f32<64><<<gridGemmO, BLK, 0, stream>>>(h, W3l, xl3, N);
  gemm_wmma_f32<64><<<gridGemmO, BLK, 0, stream>>>(h, W3r, xr3, N);
  init_layer<<<gridElemO, BLK, 0, stream>>>(mmax, den, lv, b3, N, O);
  edge_scores<64><<<gridWaveE, BLK, 0, stream>>>(xl3, xr3, a3, ei, E, T, ebuf, mmax);
  edge_exp<<<gridEdge, BLK, 0, stream>>>(ei, E, T, ebuf, mmax, den);
  edge_aggr<64><<<gridWaveE, BLK, 0, stream>>>(xl3, ei, E, T, ebuf, den, lv);
}